// DualGCN_16372415332644
// MI455X (gfx1250) — compile-verified
//
#include <hip/hip_runtime.h>
#include <hip/hip_bf16.h>

typedef float v2f __attribute__((ext_vector_type(2)));
typedef float v8f __attribute__((ext_vector_type(8)));

#define N_NODES  100000
#define N_EDGES  3200000
#define N_GRAPHS 64
#define IN_F     11
#define HID      32

// ---------------------------------------------------------------------------
// ws layout (byte offsets, 256-aligned):
//   deg/dinv : float[N_NODES]
//   H        : float[N_NODES*HID]   (x @ W_gcn)
//   agg      : float[N_NODES*HID]   (scatter-add target)
//   pooled   : float[N_GRAPHS*HID]
//   cnt      : float[N_GRAPHS]
// ---------------------------------------------------------------------------

__global__ void init_ws_k(float* __restrict__ deg, float* __restrict__ agg,
                          float* __restrict__ pooled, float* __restrict__ cnt) {
  const int i = blockIdx.x * blockDim.x + threadIdx.x;   // covers N_NODES*HID
  agg[i] = 0.0f;
  if (i < N_NODES)        deg[i]    = 1.0f;              // self-loop counted
  if (i < N_GRAPHS * HID) pooled[i] = 0.0f;
  if (i < N_GRAPHS)       cnt[i]    = 0.0f;
}

__global__ void degree_k(const long long* __restrict__ dst, float* __restrict__ deg) {
  const int e = blockIdx.x * blockDim.x + threadIdx.x;
  if (e < N_EDGES) atomicAdd(&deg[(int)dst[e]], 1.0f);
}

__global__ void dinv_k(float* __restrict__ deg) {
  const int i = blockIdx.x * blockDim.x + threadIdx.x;
  if (i < N_NODES) deg[i] = rsqrtf(deg[i]);              // deg >= 1 always
}

// H = X @ W  via v_wmma_f32_16x16x4_f32; one wave -> 16 rows x 32 cols, 6 WMMAs
__global__ __launch_bounds__(256) void gemm_wmma_k(const float* __restrict__ X,
                                                   const float* __restrict__ W,
                                                   float* __restrict__ H) {
  const int wave = blockIdx.x * 8 + (threadIdx.x >> 5);
  const int lane = threadIdx.x & 31;
  const int numTiles = (N_NODES + 15) / 16;
  if (wave >= numTiles) return;                           // wave-uniform: EXEC stays all-1

  const int row  = wave * 16 + (lane & 15);               // A row for this lane
  const int koff = (lane < 16) ? 0 : 2;                   // K sub-offset per lane half
  const int col  = lane & 15;                             // B/D column within tile

  v8f c0 = {};                                            // cols 0..15
  v8f c1 = {};                                            // cols 16..31
#pragma unroll
  for (int kb = 0; kb < 3; ++kb) {                        // K = 11 padded to 12
    const int k0 = kb * 4 + koff;
    v2f a, b0, b1;
    const bool rok = (row < N_NODES);
    a.x  = (rok && k0     < IN_F) ? X[row * IN_F + k0]     : 0.0f;
    a.y  = (rok && k0 + 1 < IN_F) ? X[row * IN_F + k0 + 1] : 0.0f;
    b0.x = (k0     < IN_F) ? W[ k0      * HID + col]       : 0.0f;
    b0.y = (k0 + 1 < IN_F) ? W[(k0 + 1) * HID + col]       : 0.0f;
    b1.x = (k0     < IN_F) ? W[ k0      * HID + 16 + col]  : 0.0f;
    b1.y = (k0 + 1 < IN_F) ? W[(k0 + 1) * HID + 16 + col]  : 0.0f;
    c0 = __builtin_amdgcn_wmma_f32_16x16x4_f32(false, a, false, b0, (short)0, c0, false, false);
    c1 = __builtin_amdgcn_wmma_f32_16x16x4_f32(false, a, false, b1, (short)0, c1, false, false);
  }

  const int rbase = wave * 16 + ((lane >= 16) ? 8 : 0);   // C/D layout: M = v (+8 upper half)
#pragma unroll
  for (int v = 0; v < 8; ++v) {
    const int r = rbase + v;
    if (r < N_NODES) {
      H[r * HID + col]      = c0[v];
      H[r * HID + 16 + col] = c1[v];
    }
  }
}

// 8 sub-threads per edge, float4 each: agg[dst] += H[src] * dinv[src]*dinv[dst]
__global__ void scatter_k(const long long* __restrict__ src,
                          const long long* __restrict__ dst,
                          const float* __restrict__ dinv,
                          const float* __restrict__ H,
                          float* __restrict__ agg) {
  const long long t = (long long)blockIdx.x * blockDim.x + threadIdx.x;
  if (t >= (long long)N_EDGES * 8) return;
  const int e = (int)(t >> 3);
  const int q = (int)(t & 7);
  const int s = (int)src[e];
  const int d = (int)dst[e];
  const float norm = dinv[s] * dinv[d];
  const float4 hv = *reinterpret_cast<const float4*>(&H[s * HID + q * 4]);
  float* a = &agg[d * HID + q * 4];
  atomicAdd(a + 0, hv.x * norm);
  atomicAdd(a + 1, hv.y * norm);
  atomicAdd(a + 2, hv.z * norm);
  atomicAdd(a + 3, hv.w * norm);
}

// relu(agg + self-loop + bias), segment-sum into LDS bins, flush to global
#define NODES_PER_BLOCK 1024
__global__ __launch_bounds__(256) void relu_pool_k(const float* __restrict__ agg,
                                                   const float* __restrict__ H,
                                                   const float* __restrict__ dinv,
                                                   const float* __restrict__ b_gcn,
                                                   const long long* __restrict__ batch,
                                                   float* __restrict__ pooled,
                                                   float* __restrict__ cnt) {
  __shared__ float bins[N_GRAPHS * HID];
  __shared__ float lcnt[N_GRAPHS];
  for (int i = threadIdx.x; i < N_GRAPHS * HID; i += 256) bins[i] = 0.0f;
  if (threadIdx.x < N_GRAPHS) lcnt[threadIdx.x] = 0.0f;
  __syncthreads();

  const int feat = threadIdx.x & 31;
  const int sub  = threadIdx.x >> 5;                      // 0..7 nodes in flight
  const float bias = b_gcn[feat];
  const int base = blockIdx.x * NODES_PER_BLOCK;

  for (int it = 0; it < NODES_PER_BLOCK / 8; ++it) {
    const int node = base + it * 8 + sub;
    if (node < N_NODES) {
      const float di = dinv[node];
      float v = agg[node * HID + feat] + H[node * HID + feat] * (di * di) + bias;
      v = fmaxf(v, 0.0f);
      const int g = (int)batch[node];
      atomicAdd(&bins[g * HID + feat], v);                // ds_add_f32 (LDS atomic)
      if (feat == 0) atomicAdd(&lcnt[g], 1.0f);
    }
  }
  __syncthreads();

  for (int i = threadIdx.x; i < N_GRAPHS * HID; i += 256)
    if (bins[i] != 0.0f) atomicAdd(&pooled[i], bins[i]);
  if (threadIdx.x < N_GRAPHS && lcnt[threadIdx.x] != 0.0f)
    atomicAdd(&cnt[threadIdx.x], lcnt[threadIdx.x]);
}

// out[g] = (sum_j pooled[g][j] * W_out[j]) / cnt[g] + b_out
__global__ void final_k(const float* __restrict__ pooled, const float* __restrict__ cnt,
                        const float* __restrict__ W_out, const float* __restrict__ b_out,
                        float* __restrict__ out) {
  const int g = threadIdx.x;
  if (g < N_GRAPHS) {
    float s = 0.0f;
#pragma unroll
    for (int j = 0; j < HID; ++j) s += pooled[g * HID + j] * W_out[j];
    out[g] = s / fmaxf(cnt[g], 1.0f) + b_out[0];
  }
}

extern "C" void kernel_launch(void* const* d_in, const int* in_sizes, int n_in,
                              void* d_out, int out_size, void* d_ws, size_t ws_size,
                              hipStream_t stream) {
  (void)in_sizes; (void)n_in; (void)out_size; (void)ws_size;
  const float*     x     = (const float*)d_in[0];
  const long long* ei    = (const long long*)d_in[1];     // [2, E] int64
  const long long* batch = (const long long*)d_in[2];     // [N] int64 (sorted)
  const float*     W_gcn = (const float*)d_in[3];
  const float*     b_gcn = (const float*)d_in[4];
  const float*     W_out = (const float*)d_in[5];
  const float*     b_out = (const float*)d_in[6];
  float* out = (float*)d_out;

  const long long* e_src = ei;
  const long long* e_dst = ei + N_EDGES;

  auto align256 = [](size_t v) { return (v + 255) & ~(size_t)255; };
  char* ws = (char*)d_ws;
  size_t off = 0;
  float* deg    = (float*)(ws + off); off = align256(off + sizeof(float) * N_NODES);
  float* H      = (float*)(ws + off); off = align256(off + sizeof(float) * (size_t)N_NODES * HID);
  float* agg    = (float*)(ws + off); off = align256(off + sizeof(float) * (size_t)N_NODES * HID);
  float* pooled = (float*)(ws + off); off = align256(off + sizeof(float) * N_GRAPHS * HID);
  float* cnt    = (float*)(ws + off); off = align256(off + sizeof(float) * N_GRAPHS);

  // 1) zero/seed workspace (agg span == N*HID == 3.2M, exactly 12500*256)
  init_ws_k<<<(N_NODES * HID) / 256, 256, 0, stream>>>(deg, agg, pooled, cnt);
  // 2) in-degree (incl. self-loop seed)
  degree_k<<<(N_EDGES + 255) / 256, 256, 0, stream>>>(e_dst, deg);
  // 3) dinv = rsqrt(deg)
  dinv_k<<<(N_NODES + 255) / 256, 256, 0, stream>>>(deg);
  // 4) H = x @ W_gcn via WMMA (6250 row-tiles, 8 waves/block)
  {
    const int tiles = (N_NODES + 15) / 16;
    gemm_wmma_k<<<(tiles + 7) / 8, 256, 0, stream>>>(x, W_gcn, H);
  }
  // 5) edge scatter-add
  {
    const long long work = (long long)N_EDGES * 8;
    scatter_k<<<(unsigned)((work + 255) / 256), 256, 0, stream>>>(e_src, e_dst, deg, H, agg);
  }
  // 6) relu + segment mean-pool partials
  relu_pool_k<<<(N_NODES + NODES_PER_BLOCK - 1) / NODES_PER_BLOCK, 256, 0, stream>>>(
      agg, H, deg, b_gcn, batch, pooled, cnt);
  // 7) final projection
  final_k<<<1, 64, 0, stream>>>(pooled, cnt, W_out, b_out, out);
}